// DynamicGCN_81123342287579
// MI455X (gfx1250) — compile-verified
//
#include <hip/hip_runtime.h>
#include <hip/hip_bf16.h>

#define DCH 128  // channels

typedef __attribute__((ext_vector_type(2))) float v2f;
typedef __attribute__((ext_vector_type(8))) float v8f;

// ---------------------------------------------------------------------------
// Degree / normalization kernels
// ---------------------------------------------------------------------------
__global__ void gcn_deg_init(float* __restrict__ deg, int n) {
    int i = blockIdx.x * blockDim.x + threadIdx.x;
    if (i < n) deg[i] = 1.0f;  // self-loop contributes 1 to every node's degree
}

__global__ void gcn_deg_accum(const int* __restrict__ dst, float* __restrict__ deg, int e) {
    int i = blockIdx.x * blockDim.x + threadIdx.x;
    if (i < e) atomicAdd(&deg[dst[i]], 1.0f);
}

__global__ void gcn_deg_rsqrt(float* __restrict__ deg, int n) {
    int i = blockIdx.x * blockDim.x + threadIdx.x;
    if (i < n) deg[i] = rsqrtf(deg[i]);  // deg >= 1 always (self-loop)
}

// ---------------------------------------------------------------------------
// Zero a float buffer (float4 stores)
// ---------------------------------------------------------------------------
__global__ void gcn_zero_f4(float4* __restrict__ p, int n4) {
    int i = blockIdx.x * blockDim.x + threadIdx.x;
    if (i < n4) p[i] = make_float4(0.f, 0.f, 0.f, 0.f);
}

// ---------------------------------------------------------------------------
// H = X @ W  (X: [n,128] row-major, W: [128,128] row-major) via f32 WMMA.
// Block = 256 threads = 8 waves. Wave w owns output cols [16w,16w+16).
// Each wave computes 4 row-tiles (64 rows per block), K stepped by 4.
//
// 32-bit A 16x4 layout:  lane<16 holds row M=lane, K = k0+0 / k0+1 in a.x/a.y
//                        lane>=16 holds row M=lane-16, K = k0+2 / k0+3
// 32-bit B 4x16 layout:  lanes 0-15 hold K rows k0+0,k0+1; lanes 16-31 hold
//                        K rows k0+2,k0+3; N = lane&15 per VGPR.
// C/D 16x16 f32: acc[i]: lanes 0-15 -> M=i, lanes 16-31 -> M=i+8; N=lane&15.
//
// Epilogue guards are per-TILE scalar branches (wave-uniform), so the common
// path stores with no EXEC-mask manipulation; a per-element fallback handles
// a partially valid tile (cold; unused when n_nodes % 16 == 0).
// ---------------------------------------------------------------------------
__global__ __launch_bounds__(256) void gcn_gemm_wmma(
    const float* __restrict__ X, const float* __restrict__ Wm,
    float* __restrict__ H, int n_nodes)
{
    const int lane = threadIdx.x & 31;
    const int wave = threadIdx.x >> 5;   // 0..7
    const int n0   = wave << 4;
    const int l15  = lane & 15;
    const int half = lane >> 4;          // 0 or 1
    const int m_base = blockIdx.x * 64;

    int row[4];
#pragma unroll
    for (int t = 0; t < 4; ++t) {
        int r = m_base + t * 16 + l15;
        row[t] = (r < n_nodes) ? r : (n_nodes - 1);  // clamp: EXEC must stay all-1s
    }

    v8f acc[4] = {{}, {}, {}, {}};

#pragma unroll 4
    for (int k0 = 0; k0 < DCH; k0 += 4) {
        const int kk = k0 + (half << 1);
        v2f b;
        b.x = Wm[kk * DCH + n0 + l15];
        b.y = Wm[(kk + 1) * DCH + n0 + l15];
        v2f a0 = *(const v2f*)(X + (size_t)row[0] * DCH + kk);
        v2f a1 = *(const v2f*)(X + (size_t)row[1] * DCH + kk);
        v2f a2 = *(const v2f*)(X + (size_t)row[2] * DCH + kk);
        v2f a3 = *(const v2f*)(X + (size_t)row[3] * DCH + kk);
        acc[0] = __builtin_amdgcn_wmma_f32_16x16x4_f32(false, a0, false, b, (short)0, acc[0], false, false);
        acc[1] = __builtin_amdgcn_wmma_f32_16x16x4_f32(false, a1, false, b, (short)0, acc[1], false, false);
        acc[2] = __builtin_amdgcn_wmma_f32_16x16x4_f32(false, a2, false, b, (short)0, acc[2], false, false);
        acc[3] = __builtin_amdgcn_wmma_f32_16x16x4_f32(false, a3, false, b, (short)0, acc[3], false, false);
    }

    const int ncol = n0 + l15;
#pragma unroll
    for (int t = 0; t < 4; ++t) {
        const int mt = m_base + t * 16;
        if (mt + 15 < n_nodes) {
            // Fast path: whole 16-row tile valid -> unpredicated stores.
#pragma unroll
            for (int i = 0; i < 8; ++i)
                H[(size_t)(mt + half * 8 + i) * DCH + ncol] = acc[t][i];
        } else if (mt < n_nodes) {
            // Cold path: partial tile (only if n_nodes % 16 != 0).
#pragma unroll
            for (int i = 0; i < 8; ++i) {
                int m = mt + half * 8 + i;
                if (m < n_nodes) H[(size_t)m * DCH + ncol] = acc[t][i];
            }
        }
    }
}

// ---------------------------------------------------------------------------
// Edge scatter: one wave32 per edge; lane moves float4 (16B) of the message.
// agg[dst] += dis[src]*dis[dst] * H[src]
// ---------------------------------------------------------------------------
__global__ __launch_bounds__(256) void gcn_edge_scatter(
    const float* __restrict__ H,
    const int* __restrict__ src, const int* __restrict__ dst,
    const float* __restrict__ dis,
    float* __restrict__ agg, int n_edges)
{
    const int warp = (blockIdx.x * blockDim.x + threadIdx.x) >> 5;
    const int lane = threadIdx.x & 31;
    if (warp >= n_edges) return;

    const int s = src[warp];
    const int d = dst[warp];
    const float nrm = dis[s] * dis[d];

    const float4 hv = *(const float4*)(H + (size_t)s * DCH + lane * 4);
    float* ap = agg + (size_t)d * DCH + lane * 4;
    atomicAdd(ap + 0, hv.x * nrm);
    atomicAdd(ap + 1, hv.y * nrm);
    atomicAdd(ap + 2, hv.z * nrm);
    atomicAdd(ap + 3, hv.w * nrm);
}

// ---------------------------------------------------------------------------
// Finalize: out = ELU(agg + H*dis^2 + bias)   (self-loop fused; elementwise)
// out may alias agg (in-place read-then-write per element).
// ---------------------------------------------------------------------------
__global__ void gcn_finalize(
    const float* __restrict__ H, const float* __restrict__ dis,
    const float* __restrict__ bias, const float* __restrict__ agg,
    float* __restrict__ out, int n_nodes)
{
    int idx = blockIdx.x * blockDim.x + threadIdx.x;
    int total = n_nodes * DCH;
    if (idx >= total) return;
    int node = idx >> 7;        // / 128
    int c    = idx & (DCH - 1);
    float ds = dis[node];
    float v  = agg[idx] + H[idx] * ds * ds + bias[c];
    out[idx] = (v > 0.0f) ? v : expm1f(v);
}

// ---------------------------------------------------------------------------
// Launch
// ---------------------------------------------------------------------------
extern "C" void kernel_launch(void* const* d_in, const int* in_sizes, int n_in,
                              void* d_out, int out_size, void* d_ws, size_t ws_size,
                              hipStream_t stream) {
    const float* x   = (const float*)d_in[0];
    const int*   ei  = (const int*)d_in[1];
    const float* W1  = (const float*)d_in[2];
    const float* b1  = (const float*)d_in[3];
    const float* W2  = (const float*)d_in[4];
    const float* b2  = (const float*)d_in[5];

    const int n_nodes = in_sizes[0] / DCH;
    const int n_edges = in_sizes[1] / 2;
    const int* src = ei;
    const int* dst = ei + n_edges;

    // Workspace layout: dis | H | AGG
    float* dis = (float*)d_ws;
    float* H   = (float*)((char*)d_ws + ((size_t)1 << 20));
    float* AGG = (float*)((char*)d_ws + ((size_t)1 << 20) + (size_t)n_nodes * DCH * sizeof(float));
    float* OUT = (float*)d_out;

    const int total   = n_nodes * DCH;
    const int blkN    = (n_nodes + 255) / 256;
    const int blkE    = (n_edges + 255) / 256;
    const int blkEw   = (n_edges + 7) / 8;          // 8 warps per 256-thread block, 1 edge/warp
    const int blkZ    = (total / 4 + 255) / 256;
    const int blkF    = (total + 255) / 256;
    const int blkGemm = (n_nodes + 63) / 64;

    // --- normalization: dis = rsqrt(deg), deg = 1 + #incoming edges ---
    gcn_deg_init  <<<blkN, 256, 0, stream>>>(dis, n_nodes);
    gcn_deg_accum <<<blkE, 256, 0, stream>>>(dst, dis, n_edges);
    gcn_deg_rsqrt <<<blkN, 256, 0, stream>>>(dis, n_nodes);

    // --- layer 1 ---
    gcn_gemm_wmma   <<<blkGemm, 256, 0, stream>>>(x, W1, H, n_nodes);
    gcn_zero_f4     <<<blkZ, 256, 0, stream>>>((float4*)AGG, total / 4);
    gcn_edge_scatter<<<blkEw, 256, 0, stream>>>(H, src, dst, dis, AGG, n_edges);
    gcn_finalize    <<<blkF, 256, 0, stream>>>(H, dis, b1, AGG, AGG, n_nodes); // out1 -> AGG

    // --- layer 2 ---
    gcn_gemm_wmma   <<<blkGemm, 256, 0, stream>>>(AGG, W2, H, n_nodes);
    gcn_zero_f4     <<<blkZ, 256, 0, stream>>>((float4*)AGG, total / 4);       // out1 consumed
    gcn_edge_scatter<<<blkEw, 256, 0, stream>>>(H, src, dst, dis, AGG, n_edges);
    gcn_finalize    <<<blkF, 256, 0, stream>>>(H, dis, b2, AGG, OUT, n_nodes);
}